// MixtureOfExperts_37864431682615
// MI455X (gfx1250) — compile-verified
//
#include <hip/hip_runtime.h>
#include <hip/hip_bf16.h>
#include <stdint.h>

#define N_TOK 4096
#define C_DIM 1024
#define H_DIM 4096
#define N_EXP 8

typedef float v2f __attribute__((ext_vector_type(2)));
typedef float v8f __attribute__((ext_vector_type(8)));

static __device__ inline v8f vzero8() {
  v8f z;
#pragma unroll
  for (int i = 0; i < 8; ++i) z[i] = 0.f;
  return z;
}

// ---------------------------------------------------------------------------
// Stage 0: zero the per-expert counters (graph-capture-safe, no memset node)
// ---------------------------------------------------------------------------
__global__ void moe_zero_counts(int* __restrict__ counts) {
  if (threadIdx.x < N_EXP) counts[threadIdx.x] = 0;
}

// ---------------------------------------------------------------------------
// Stage 1: router. One wave32 per token: logits over E=8 experts (Wg staged
// in LDS), butterfly reduce, top-2 + renormalize (softmax denominator cancels),
// build per-expert token lists with atomics. Entry encoding: token*2 + slot.
// ---------------------------------------------------------------------------
__global__ __launch_bounds__(256) void moe_router(
    const float* __restrict__ x, const float* __restrict__ Wg,
    int* __restrict__ counts, int* __restrict__ lists, float* __restrict__ wts) {
  __shared__ __align__(16) float sWg[N_EXP * C_DIM];
  const int tid = threadIdx.x;
  for (int i = tid; i < N_EXP * C_DIM; i += 256) sWg[i] = Wg[i];
  __syncthreads();

  const int lane = tid & 31;
  const int wv = tid >> 5;
  const int t = blockIdx.x * 8 + wv;
  if (t >= N_TOK) return;

  float acc[N_EXP];
#pragma unroll
  for (int e = 0; e < N_EXP; ++e) acc[e] = 0.f;

  const float* xr = x + (size_t)t * C_DIM;
  for (int j = 0; j < C_DIM / 32; ++j) {
    const int col = j * 32 + lane;
    const float v = xr[col];
#pragma unroll
    for (int e = 0; e < N_EXP; ++e) acc[e] += v * sWg[e * C_DIM + col];
  }
  // wave32 butterfly reduction: every lane ends with the full 8 logits
#pragma unroll
  for (int off = 16; off > 0; off >>= 1) {
#pragma unroll
    for (int e = 0; e < N_EXP; ++e) acc[e] += __shfl_xor(acc[e], off, 32);
  }

  float mx = acc[0];
#pragma unroll
  for (int e = 1; e < N_EXP; ++e) mx = fmaxf(mx, acc[e]);
  float p[N_EXP];
#pragma unroll
  for (int e = 0; e < N_EXP; ++e) p[e] = __expf(acc[e] - mx);
  // top-2 (first index wins ties, matching jax.lax.top_k)
  int s0 = 0;
#pragma unroll
  for (int e = 1; e < N_EXP; ++e) if (p[e] > p[s0]) s0 = e;
  int s1 = (s0 == 0) ? 1 : 0;
#pragma unroll
  for (int e = 0; e < N_EXP; ++e) if (e != s0 && p[e] > p[s1]) s1 = e;
  const float denom = p[s0] + p[s1];  // softmax sum cancels in renormalization

  if (lane == 0) {
    int pos0 = atomicAdd(&counts[s0], 1);
    lists[s0 * N_TOK + pos0] = t * 2 + 0;
    wts[t * 2 + 0] = p[s0] / denom;
    int pos1 = atomicAdd(&counts[s1], 1);
    lists[s1 * N_TOK + pos1] = t * 2 + 1;
    wts[t * 2 + 1] = p[s1] / denom;
  }
}

// ---------------------------------------------------------------------------
// GEMM tiling shared constants.
//   A tile: 128 rows x 32 K, row-major, stride 36 floats (144 B, 16B-aligned).
//   B tile: K-pair interleaved: element (k, n) -> Bp[k>>1][n*2 + (k&1)],
//           kp-row stride 288 floats (256 data + 32 pad). Each WMMA B frag is
//           then ONE aligned ds_load_b64 straight into the operand VGPR pair;
//           +288 dwords shifts LDS banks by 32, so the khalf=0 and khalf=2
//           half-waves read disjoint bank windows (conflict-free).
// ---------------------------------------------------------------------------
#define APAD 36
#define BRS 288

// ---------------------------------------------------------------------------
// Stage 2: per-expert GEMM1 + fused SwiGLU using V_WMMA_F32_16X16X4_F32.
// Block = 256 threads (8 waves), tile 128(M) x 128(N) x 32(K). Wave w owns
// output rows [16w,16w+16). A rows are token rows gathered via the expert
// list; B tiles come from W1[e]/W3[e] ([C,H] row-major, coalesced).
// ---------------------------------------------------------------------------
__global__ __launch_bounds__(256) void moe_gemm1(
    const float* __restrict__ x, const float* __restrict__ W1,
    const float* __restrict__ W3, const int* __restrict__ counts,
    const int* __restrict__ lists, float* __restrict__ g) {
  __shared__ __align__(16) float As[128 * APAD];
  __shared__ __align__(16) float B1s[16 * BRS];
  __shared__ __align__(16) float B3s[16 * BRS];
  __shared__ int ents[128];

  const int e = blockIdx.z;
  const int cnt = counts[e];
  const int m0 = blockIdx.y * 128;
  if (m0 >= cnt) return;  // predicate fixed grid on dynamic routing counts
  const int n0 = blockIdx.x * 128;

  const int tid = threadIdx.x, lane = tid & 31, wv = tid >> 5;
  if (tid < 128) ents[tid] = (m0 + tid < cnt) ? lists[e * N_TOK + m0 + tid] : -1;

  const float* W1e = W1 + (size_t)e * C_DIM * H_DIM;
  const float* W3e = W3 + (size_t)e * C_DIM * H_DIM;

  v8f acc1[8], acc3[8];
#pragma unroll
  for (int nt = 0; nt < 8; ++nt) { acc1[nt] = vzero8(); acc3[nt] = vzero8(); }

  const int arow = tid >> 1, ac0 = (tid & 1) * 16;  // A: 2 threads/row
  const int bkp = tid >> 4, bc0 = (tid & 15) * 8;   // B: 16 threads/K-pair-row

  for (int k0 = 0; k0 < C_DIM; k0 += 32) {
    __syncthreads();
    {  // A tile: gathered token rows (fp32, contiguous along K=C)
      const int entv = ents[arow];
      if (entv >= 0) {
        const float4* src =
            (const float4*)(x + (size_t)(entv >> 1) * C_DIM + k0 + ac0);
#pragma unroll
        for (int q = 0; q < 4; ++q)
          *(float4*)&As[arow * APAD + ac0 + q * 4] = src[q];
      } else {
        const float4 z = {0.f, 0.f, 0.f, 0.f};
#pragma unroll
        for (int q = 0; q < 4; ++q)
          *(float4*)&As[arow * APAD + ac0 + q * 4] = z;
      }
    }
    {  // W1/W3 tiles: two adjacent K rows per thread, pair-interleaved store
      const float* r10 = W1e + (size_t)(k0 + 2 * bkp) * H_DIM + n0 + bc0;
      const float* r30 = W3e + (size_t)(k0 + 2 * bkp) * H_DIM + n0 + bc0;
      float4 a0 = ((const float4*)r10)[0], a1 = ((const float4*)r10)[1];
      float4 b0 = ((const float4*)(r10 + H_DIM))[0],
             b1 = ((const float4*)(r10 + H_DIM))[1];
      float4 c0v = ((const float4*)r30)[0], c1 = ((const float4*)r30)[1];
      float4 d0 = ((const float4*)(r30 + H_DIM))[0],
             d1 = ((const float4*)(r30 + H_DIM))[1];
      const float ax[8] = {a0.x, a0.y, a0.z, a0.w, a1.x, a1.y, a1.z, a1.w};
      const float bx[8] = {b0.x, b0.y, b0.z, b0.w, b1.x, b1.y, b1.z, b1.w};
      const float cx[8] = {c0v.x, c0v.y, c0v.z, c0v.w, c1.x, c1.y, c1.z, c1.w};
      const float dx[8] = {d0.x, d0.y, d0.z, d0.w, d1.x, d1.y, d1.z, d1.w};
#pragma unroll
      for (int j = 0; j < 8; ++j) {
        v2f p1, p3;
        p1.x = ax[j]; p1.y = bx[j];
        p3.x = cx[j]; p3.y = dx[j];
        *(v2f*)&B1s[bkp * BRS + (bc0 + j) * 2] = p1;
        *(v2f*)&B3s[bkp * BRS + (bc0 + j) * 2] = p3;
      }
      if (k0 + 32 < C_DIM) {  // global_prefetch_b8 on next weight tiles
        __builtin_prefetch(W1e + (size_t)(k0 + 32 + 2 * bkp) * H_DIM + n0 + bc0,
                           0, 1);
        __builtin_prefetch(W3e + (size_t)(k0 + 32 + 2 * bkp) * H_DIM + n0 + bc0,
                           0, 1);
      }
    }
    __syncthreads();

    const int mrow = (wv << 4) + (lane & 15);
    const int khalf = (lane >> 4) << 1;  // lanes 0-15 -> K(0,1); 16-31 -> K(2,3)
#pragma unroll
    for (int ks = 0; ks < 32; ks += 4) {
      v2f a;
      a.x = As[mrow * APAD + ks + khalf];
      a.y = As[mrow * APAD + ks + khalf + 1];
      const int kp = (ks + khalf) >> 1;
#pragma unroll
      for (int nt = 0; nt < 8; ++nt) {
        const int c = nt * 16 + (lane & 15);
        const v2f b1 = *(const v2f*)&B1s[kp * BRS + c * 2];
        const v2f b3 = *(const v2f*)&B3s[kp * BRS + c * 2];
        acc1[nt] = __builtin_amdgcn_wmma_f32_16x16x4_f32(
            false, a, false, b1, (short)0, acc1[nt], false, false);
        acc3[nt] = __builtin_amdgcn_wmma_f32_16x16x4_f32(
            false, a, false, b3, (short)0, acc3[nt], false, false);
      }
    }
  }

  // Epilogue: g[entry] = silu(h1) * h3, rows scattered by slot-entry index.
  const int cbase = lane & 15;
  const int mhi = (lane >> 4) << 3;  // D layout: VGPR r -> M=r / M=8+r
#pragma unroll
  for (int nt = 0; nt < 8; ++nt) {
#pragma unroll
    for (int r = 0; r < 8; ++r) {
      const int rowIdx = (wv << 4) + mhi + r;
      const int entv = ents[rowIdx];
      if (entv >= 0) {
        const float h1 = acc1[nt][r], h3 = acc3[nt][r];
        const float sig = 1.f / (1.f + __expf(-h1));
        g[(size_t)entv * H_DIM + n0 + nt * 16 + cbase] = h1 * sig * h3;
      }
    }
  }
}

// ---------------------------------------------------------------------------
// Stage 3: per-expert GEMM2: g rows x W2[e] ([H,C] row-major) -> obuf[entry].
// Same tiling; K = H = 4096 (128 iterations).
// ---------------------------------------------------------------------------
__global__ __launch_bounds__(256) void moe_gemm2(
    const float* __restrict__ g, const float* __restrict__ W2,
    const int* __restrict__ counts, const int* __restrict__ lists,
    float* __restrict__ obuf) {
  __shared__ __align__(16) float As[128 * APAD];
  __shared__ __align__(16) float B2s[16 * BRS];
  __shared__ int ents[128];

  const int e = blockIdx.z;
  const int cnt = counts[e];
  const int m0 = blockIdx.y * 128;
  if (m0 >= cnt) return;
  const int n0 = blockIdx.x * 128;

  const int tid = threadIdx.x, lane = tid & 31, wv = tid >> 5;
  if (tid < 128) ents[tid] = (m0 + tid < cnt) ? lists[e * N_TOK + m0 + tid] : -1;

  const float* W2e = W2 + (size_t)e * H_DIM * C_DIM;

  v8f acc[8];
#pragma unroll
  for (int nt = 0; nt < 8; ++nt) acc[nt] = vzero8();

  const int arow = tid >> 1, ac0 = (tid & 1) * 16;
  const int bkp = tid >> 4, bc0 = (tid & 15) * 8;

  for (int k0 = 0; k0 < H_DIM; k0 += 32) {
    __syncthreads();
    {
      const int entv = ents[arow];
      if (entv >= 0) {
        const float4* src =
            (const float4*)(g + (size_t)entv * H_DIM + k0 + ac0);
#pragma unroll
        for (int q = 0; q < 4; ++q)
          *(float4*)&As[arow * APAD + ac0 + q * 4] = src[q];
      } else {
        const float4 z = {0.f, 0.f, 0.f, 0.f};
#pragma unroll
        for (int q = 0; q < 4; ++q)
          *(float4*)&As[arow * APAD + ac0 + q * 4] = z;
      }
    }
    {
      const float* r20 = W2e + (size_t)(k0 + 2 * bkp) * C_DIM + n0 + bc0;
      float4 a0 = ((const float4*)r20)[0], a1 = ((const float4*)r20)[1];
      float4 b0 = ((const float4*)(r20 + C_DIM))[0],
             b1 = ((const float4*)(r20 + C_DIM))[1];
      const float ax[8] = {a0.x, a0.y, a0.z, a0.w, a1.x, a1.y, a1.z, a1.w};
      const float bx[8] = {b0.x, b0.y, b0.z, b0.w, b1.x, b1.y, b1.z, b1.w};
#pragma unroll
      for (int j = 0; j < 8; ++j) {
        v2f p2;
        p2.x = ax[j]; p2.y = bx[j];
        *(v2f*)&B2s[bkp * BRS + (bc0 + j) * 2] = p2;
      }
      if (k0 + 32 < H_DIM)
        __builtin_prefetch(W2e + (size_t)(k0 + 32 + 2 * bkp) * C_DIM + n0 + bc0,
                           0, 1);
    }
    __syncthreads();

    const int mrow = (wv << 4) + (lane & 15);
    const int khalf = (lane >> 4) << 1;
#pragma unroll
    for (int ks = 0; ks < 32; ks += 4) {
      v2f a;
      a.x = As[mrow * APAD + ks + khalf];
      a.y = As[mrow * APAD + ks + khalf + 1];
      const int kp = (ks + khalf) >> 1;
#pragma unroll
      for (int nt = 0; nt < 8; ++nt) {
        const int c = nt * 16 + (lane & 15);
        const v2f b = *(const v2f*)&B2s[kp * BRS + c * 2];
        acc[nt] = __builtin_amdgcn_wmma_f32_16x16x4_f32(
            false, a, false, b, (short)0, acc[nt], false, false);
      }
    }
  }

  const int cbase = lane & 15;
  const int mhi = (lane >> 4) << 3;
#pragma unroll
  for (int nt = 0; nt < 8; ++nt) {
#pragma unroll
    for (int r = 0; r < 8; ++r) {
      const int rowIdx = (wv << 4) + mhi + r;
      const int entv = ents[rowIdx];
      if (entv >= 0)
        obuf[(size_t)entv * C_DIM + n0 + nt * 16 + cbase] = acc[nt][r];
    }
  }
}

// ---------------------------------------------------------------------------
// Stage 4: deterministic combine: out[t] = w0*obuf[2t] + w1*obuf[2t+1].
// ---------------------------------------------------------------------------
__global__ __launch_bounds__(256) void moe_combine(
    const float* __restrict__ obuf, const float* __restrict__ wts,
    float* __restrict__ out) {
  const int idx = blockIdx.x * 256 + threadIdx.x;  // N_TOK * C_DIM elements
  const int t = idx >> 10;
  const int c = idx & (C_DIM - 1);
  out[idx] = wts[2 * t] * obuf[(size_t)(2 * t) * C_DIM + c] +
             wts[2 * t + 1] * obuf[(size_t)(2 * t + 1) * C_DIM + c];
}

// ---------------------------------------------------------------------------
extern "C" void kernel_launch(void* const* d_in, const int* in_sizes, int n_in,
                              void* d_out, int out_size, void* d_ws,
                              size_t ws_size, hipStream_t stream) {
  (void)in_sizes; (void)n_in; (void)out_size; (void)ws_size;
  const float* x = (const float*)d_in[0];    // [N, C]
  const float* Wg = (const float*)d_in[1];   // [E, C]
  const float* W1 = (const float*)d_in[2];   // [E, C, H]
  const float* W2 = (const float*)d_in[3];   // [E, H, C]
  const float* W3 = (const float*)d_in[4];   // [E, C, H]
  float* out = (float*)d_out;                // [N, C]

  // workspace carve-out (~160.2 MB total)
  char* p = (char*)d_ws;
  int* counts = (int*)p;               p += 256;
  int* lists = (int*)p;                p += (size_t)N_EXP * N_TOK * sizeof(int);
  float* wts = (float*)p;              p += (size_t)2 * N_TOK * sizeof(float);
  p = (char*)(((uintptr_t)p + 255) & ~(uintptr_t)255);
  float* g = (float*)p;                p += (size_t)2 * N_TOK * H_DIM * sizeof(float);
  float* obuf = (float*)p;             /* 2*N_TOK*C_DIM floats */

  moe_zero_counts<<<1, 32, 0, stream>>>(counts);
  moe_router<<<N_TOK / 8, 256, 0, stream>>>(x, Wg, counts, lists, wts);
  // expert-major scheduling (z slowest) keeps each 32MB expert slab hot in L2
  moe_gemm1<<<dim3(H_DIM / 128, N_TOK / 128, N_EXP), 256, 0, stream>>>(
      x, W1, W3, counts, lists, g);
  moe_gemm2<<<dim3(C_DIM / 128, N_TOK / 128, N_EXP), 256, 0, stream>>>(
      g, W2, counts, lists, obuf);
  moe_combine<<<(N_TOK * C_DIM) / 256, 256, 0, stream>>>(obuf, wts, out);
}